// CE_46686294507777
// MI455X (gfx1250) — compile-verified
//
#include <hip/hip_runtime.h>

// ---------------------------------------------------------------------------
// WENO5 conservation-law RHS:  dudt = -3*WENO(u u_x) + 0.1*u_xx - 1.0*u_xxx
// Grid: [B=4096, NX=8192] fp32, periodic.  VALU-bound stencil:
//  - tile staged into LDS via CDNA5 async global->LDS loads (ASYNCcnt path)
//  - fhat computed ONCE per interface into LDS (shared by 2 outputs)
//  - smoothness indicators + their reciprocals computed ONCE per 5-cell
//    window and shared between the left-biased (um) and right-biased (up)
//    reconstructions (mirrored betas are the same set reversed) -> ~30%
//    fewer VALU ops on the dominant phase
//  - v_rcp_f32 instead of exact divides for the self-normalizing weights
// ---------------------------------------------------------------------------

namespace {
constexpr int   kNX      = 8192;
constexpr int   kTile    = 1024;          // cells per block
constexpr int   kThreads = 256;           // 8 waves (wave32)
constexpr int   kHalo    = 3;             // WENO5 radius
constexpr int   kStage   = kTile + 2 * kHalo + 1;  // cells tile0-3 .. tile0+kTile+3
constexpr float kEps     = 1e-16f;
// DX = 16/8192 = 1/512
constexpr float kInvDx   = 512.0f;
constexpr float kInvDx2  = 512.0f * 512.0f;                 // 1/dx^2
constexpr float kInv2Dx3 = 0.5f * 512.0f * 512.0f * 512.0f; // 1/(2 dx^3)
constexpr float kAlpha   = 3.0f;
constexpr float kBeta    = 0.1f;
constexpr float kGamma   = 1.0f;
constexpr float kSixth   = 1.0f / 6.0f;
constexpr float kC13     = 13.0f / 12.0f;
}

// CDNA5 async global->LDS staging (gfx1250 only; guarded fallback).
#if defined(__gfx1250__) && __has_builtin(__builtin_amdgcn_global_load_async_to_lds_b32)
#define CE_ASYNC_LDS 1
typedef __attribute__((address_space(1))) int as1_int;
typedef __attribute__((address_space(3))) int as3_int;
#endif

__device__ __forceinline__ float fast_rcp(float x) {
#if defined(__HIP_DEVICE_COMPILE__)
  return __builtin_amdgcn_rcpf(x);   // ~1 ulp, plenty for WENO weights
#else
  return 1.0f / x;
#endif
}

// Full left-biased WENO5 reconstruction (used only for the one odd interface).
__device__ __forceinline__ float weno5(float um2, float um1, float u0,
                                       float up1, float up2) {
  const float d0 = um2 - 2.0f * um1 + u0;
  const float d1 = um1 - 2.0f * u0  + up1;
  const float d2 = u0  - 2.0f * up1 + up2;
  const float e0 = um2 - 4.0f * um1 + 3.0f * u0;
  const float e1 = um1 - up1;
  const float e2 = 3.0f * u0 - 4.0f * up1 + up2;
  const float b0 = kC13 * d0 * d0 + 0.25f * e0 * e0;
  const float b1 = kC13 * d1 * d1 + 0.25f * e1 * e1;
  const float b2 = kC13 * d2 * d2 + 0.25f * e2 * e2;
  const float r0 = fast_rcp(kEps + b0);
  const float r1 = fast_rcp(kEps + b1);
  const float r2 = fast_rcp(kEps + b2);
  const float a0 = (0.1f * r0) * r0;
  const float a1 = (0.6f * r1) * r1;
  const float a2 = (0.3f * r2) * r2;
  const float p0 = (2.0f * um2 - 7.0f * um1 + 11.0f * u0) * kSixth;
  const float p1 = (-um1 + 5.0f * u0 + 2.0f * up1) * kSixth;
  const float p2 = (2.0f * u0 + 5.0f * up1 - up2) * kSixth;
  return (a0 * p0 + a1 * p1 + a2 * p2) * fast_rcp(a0 + a1 + a2);
}

__device__ __forceinline__ float godunov_flux(float um, float up) {
  const float lm = fmaxf(um, 0.0f);
  const float rp = fminf(up, 0.0f);
  return 0.5f * fmaxf(lm * lm, rp * rp);
}

__global__ __launch_bounds__(kThreads)
void ce_dudt_weno5_kernel(const float* __restrict__ u, float* __restrict__ out) {
  __shared__ float su[kStage];      // cells  tile0-3 .. tile0+kTile+3
  __shared__ float sfh[kTile + 1];  // fhat at interfaces tile0 .. tile0+kTile

  const int    tile0   = blockIdx.x * kTile;
  const int    row     = blockIdx.y;
  const size_t rowbase = (size_t)row * kNX;
  const float* urow    = u + rowbase;

  // ---- Stage tile + halo into LDS (periodic wrap; NX is a power of two) ----
  for (int k = threadIdx.x; k < kStage; k += kThreads) {
    const int gidx = (tile0 - kHalo + k) & (kNX - 1);
#if defined(CE_ASYNC_LDS)
    __builtin_amdgcn_global_load_async_to_lds_b32(
        (as1_int*)(urow + gidx), (as3_int*)&su[k], 0, 0);
#else
    su[k] = urow[gidx];
#endif
  }
#if defined(CE_ASYNC_LDS)
#if __has_builtin(__builtin_amdgcn_s_wait_asynccnt)
  __builtin_amdgcn_s_wait_asynccnt(0);
#else
  asm volatile("s_wait_asynccnt 0" ::: "memory");
#endif
#endif
  __syncthreads();

  // ---- Godunov flux at interfaces, 4 consecutive per thread ---------------
  // su[cell + 3] holds cell (tile0 + cell).  Thread owns interfaces
  // j0..j0+3; they need cells j0-3..j0+5 == su[j0 .. j0+8] == c[0..8].
  {
    const int j0 = threadIdx.x * 4;
    float c[9];
#pragma unroll
    for (int k = 0; k < 9; ++k) c[k] = su[j0 + k];

    // Window w is centered at cell (j0-1+w), cells c[w..w+4].  Its betas are
    // shared (reversed) between up@interface(j0-1+w) and um@interface(j0+w).
    float q0[5], q1[5], q2[5];
#pragma unroll
    for (int w = 0; w < 5; ++w) {
      const float v0 = c[w], v1 = c[w + 1], v2 = c[w + 2];
      const float v3 = c[w + 3], v4 = c[w + 4];
      const float d0 = v0 - 2.0f * v1 + v2;
      const float d1 = v1 - 2.0f * v2 + v3;
      const float d2 = v2 - 2.0f * v3 + v4;
      const float e0 = v0 - 4.0f * v1 + 3.0f * v2;
      const float e1 = v1 - v3;
      const float e2 = 3.0f * v2 - 4.0f * v3 + v4;
      const float b0 = kC13 * d0 * d0 + 0.25f * e0 * e0;
      const float b1 = kC13 * d1 * d1 + 0.25f * e1 * e1;
      const float b2 = kC13 * d2 * d2 + 0.25f * e2 * e2;
      const float r0 = fast_rcp(kEps + b0);
      const float r1 = fast_rcp(kEps + b1);
      const float r2 = fast_rcp(kEps + b2);
      q0[w] = r0 * r0;   // q_i = 1/(eps+B_i)^2
      q1[w] = r1 * r1;
      q2[w] = r2 * r2;
    }

#pragma unroll
    for (int m = 0; m < 4; ++m) {
      // um at interface j0+m: window m (cells c[m..m+4]), forward weights.
      const float pa0 = (2.0f * c[m] - 7.0f * c[m + 1] + 11.0f * c[m + 2]) * kSixth;
      const float pa1 = (-c[m + 1] + 5.0f * c[m + 2] + 2.0f * c[m + 3]) * kSixth;
      const float pa2 = (2.0f * c[m + 2] + 5.0f * c[m + 3] - c[m + 4]) * kSixth;
      const float aa0 = 0.1f * q0[m];
      const float aa1 = 0.6f * q1[m];
      const float aa2 = 0.3f * q2[m];
      const float um  = (aa0 * pa0 + aa1 * pa1 + aa2 * pa2) *
                        fast_rcp(aa0 + aa1 + aa2);
      // up at interface j0+m: window m+1 (cells c[m+1..m+5]), reversed.
      const float pb0 = (2.0f * c[m + 5] - 7.0f * c[m + 4] + 11.0f * c[m + 3]) * kSixth;
      const float pb1 = (-c[m + 4] + 5.0f * c[m + 3] + 2.0f * c[m + 2]) * kSixth;
      const float pb2 = (2.0f * c[m + 3] + 5.0f * c[m + 2] - c[m + 1]) * kSixth;
      const float ab0 = 0.1f * q2[m + 1];
      const float ab1 = 0.6f * q1[m + 1];
      const float ab2 = 0.3f * q0[m + 1];
      const float up  = (ab0 * pb0 + ab1 * pb1 + ab2 * pb2) *
                        fast_rcp(ab0 + ab1 + ab2);
      sfh[j0 + m] = godunov_flux(um, up);
    }
  }

  // Odd 1025th interface (i == kTile): needs cells kTile-3..kTile+2
  // == su[kTile .. kTile+5]; all staged.  One lane does the plain path.
  if (threadIdx.x == kThreads - 1) {
    const float um = weno5(su[kTile + 0], su[kTile + 1], su[kTile + 2],
                           su[kTile + 3], su[kTile + 4]);
    const float up = weno5(su[kTile + 5], su[kTile + 4], su[kTile + 3],
                           su[kTile + 2], su[kTile + 1]);
    sfh[kTile] = godunov_flux(um, up);
  }
  __syncthreads();

  // ---- Combine: -alpha*uux + beta*uxx - gamma*uxxx; float4 stores ---------
  const int j0 = threadIdx.x * 4;
  float res[4];
#pragma unroll
  for (int m = 0; m < 4; ++m) {
    const int   j    = j0 + m;
    const float uux  = (sfh[j + 1] - sfh[j]) * kInvDx;
    const float um2  = su[j + 1];   // u[j-2]
    const float um1  = su[j + 2];   // u[j-1]
    const float u0   = su[j + 3];   // u[j]
    const float up1  = su[j + 4];   // u[j+1]
    const float up2  = su[j + 5];   // u[j+2]
    const float uxx  = (um1 - 2.0f * u0 + up1) * kInvDx2;
    const float uxxx = (up2 - 2.0f * up1 + 2.0f * um1 - um2) * kInv2Dx3;
    res[m] = -kAlpha * uux + kBeta * uxx - kGamma * uxxx;
  }
  *reinterpret_cast<float4*>(out + rowbase + tile0 + j0) =
      make_float4(res[0], res[1], res[2], res[3]);
}

extern "C" void kernel_launch(void* const* d_in, const int* in_sizes, int n_in,
                              void* d_out, int out_size, void* d_ws, size_t ws_size,
                              hipStream_t stream) {
  (void)n_in; (void)out_size; (void)d_ws; (void)ws_size;
  const float* u   = (const float*)d_in[0];   // [B, NX] fp32
  float*       out = (float*)d_out;           // [B, NX] fp32
  const int rows = in_sizes[0] / kNX;         // B = 4096
  dim3 grid(kNX / kTile, rows, 1);
  dim3 block(kThreads, 1, 1);
  hipLaunchKernelGGL(ce_dudt_weno5_kernel, grid, block, 0, stream, u, out);
}